// EdgewiseReduce_41506563948778
// MI455X (gfx1250) — compile-verified
//
#include <hip/hip_runtime.h>
#include <hip/hip_bf16.h>

// EdgewiseReduce (segment_sum of sorted edges) for MI455X / gfx1250.
//
// Bandwidth-bound op (~224 MB traffic, ~9.6us floor at 23.3 TB/s).
// Strategy: per-wave contiguous edge chunks; accumulate runs of up to 16
// destination nodes in registers using V_WMMA_F32_16X16X4_F32 with a one-hot
// A-matrix (segmented-sum-as-matmul), flush windows with global_atomic_add_f32.
// All WMMA-path branching is on readfirstlane'd scalars so EXEC == all-1s.

typedef __attribute__((ext_vector_type(2))) float v2f;
typedef __attribute__((ext_vector_type(4))) float v4f;
typedef __attribute__((ext_vector_type(8))) float v8f;

#define WAVE            32
#define WAVES_PER_BLK   8
#define BLOCK           (WAVE * WAVES_PER_BLK)
#define EDGES_PER_WAVE  128        // 32 groups of 4 edges
#define D_FEAT          64

static __device__ __forceinline__ long lmin(long a, long b) { return a < b ? a : b; }

__global__ void er_zero_kernel(float* __restrict__ out, int n4) {
    int i = blockIdx.x * blockDim.x + threadIdx.x;
    if (i < n4) {
        v4f z = {};
        ((v4f*)out)[i] = z;
    }
}

__global__ __launch_bounds__(BLOCK)
void er_wmma_scatter_kernel(const int* __restrict__ edge_center,   // [E] sorted
                            const float* __restrict__ edge_feat,   // [E, 64]
                            float* __restrict__ out,               // [N, 64] pre-zeroed
                            int E, int N)
{
    const int lane  = threadIdx.x & (WAVE - 1);
    const int wid   = blockIdx.x * WAVES_PER_BLK + (threadIdx.x >> 5);
    const long chunk = (long)wid * EDGES_PER_WAVE;
    if (chunk >= E) return;                       // whole wave exits (uniform)
    const long chunk_end = lmin((long)E, chunk + EDGES_PER_WAVE);

    // WMMA lane roles (16x16x4 f32):
    //  A (16x4): lanes 0-15 hold M=lane, K-pair {0,1}; lanes 16-31 hold K-pair {2,3}
    //  B (4x16): lanes 0-15 hold N=lane, K-pair {0,1}; lanes 16-31 K-pair {2,3}
    //  C/D (16x16): VGPR j -> row j (lanes 0-15) / row j+8 (lanes 16-31), col = lane&15
    const int m_row = lane & 15;
    const int hi    = lane >> 4;
    const int kb    = hi * 2;      // first K (edge-in-group) this lane supplies

    v8f acc0 = {}, acc1 = {}, acc2 = {}, acc3 = {};
    const v8f vz = {};
    int win_base = -1;             // node id of window row 0; -1 => accs are zero

    auto flush_win = [&]() {
#pragma unroll
        for (int j = 0; j < 8; ++j) {
            int node = win_base + j + hi * 8;
            if (node < N) {
                float* op = out + (size_t)node * D_FEAT + m_row;
                atomicAdd(op +  0, acc0[j]);
                atomicAdd(op + 16, acc1[j]);
                atomicAdd(op + 32, acc2[j]);
                atomicAdd(op + 48, acc3[j]);
            }
        }
    };

    for (long base = chunk; base < chunk_end; base += 4) {
        // Prefetch next group's features (global_prefetch_b8, speculative).
        if (base + 8 <= chunk_end)
            __builtin_prefetch(edge_feat + (size_t)(base + 4) * D_FEAT + lane * 8, 0, 0);

        const int cnt = (int)lmin((long)4, chunk_end - base);

        // Load the group's 4 destination nodes; force scalar so all window /
        // span decisions are wave-uniform (EXEC stays full for WMMA).
        int c0 = edge_center[base];
        int c1 = edge_center[lmin(base + 1, (long)E - 1)];
        int c2 = edge_center[lmin(base + 2, (long)E - 1)];
        int c3 = edge_center[lmin(base + 3, (long)E - 1)];
        c0 = __builtin_amdgcn_readfirstlane(c0);
        c1 = __builtin_amdgcn_readfirstlane(c1);
        c2 = __builtin_amdgcn_readfirstlane(c2);
        c3 = __builtin_amdgcn_readfirstlane(c3);

        if (cnt == 4) {
            if (win_base >= 0 && c3 > win_base + 15) {
                flush_win();                       // window exhausted
                win_base = -1;
                acc0 = vz; acc1 = vz; acc2 = vz; acc3 = vz;
            }
            if (win_base < 0) win_base = c0;       // open fresh window at c0

            if (c3 <= win_base + 15) {
                // --- WMMA segmented-accumulate path ---
                const int e0 = hi ? c2 : c0;       // node of K = kb
                const int e1 = hi ? c3 : c1;       // node of K = kb+1
                v2f a;
                a.x = (e0 - win_base == m_row) ? 1.0f : 0.0f;
                a.y = (e1 - win_base == m_row) ? 1.0f : 0.0f;

                const float* fb = edge_feat + (size_t)(base + kb) * D_FEAT + m_row;
                v2f b0 = { fb[ 0], fb[ 0 + D_FEAT] };
                v2f b1 = { fb[16], fb[16 + D_FEAT] };
                v2f b2 = { fb[32], fb[32 + D_FEAT] };
                v2f b3 = { fb[48], fb[48 + D_FEAT] };

                acc0 = __builtin_amdgcn_wmma_f32_16x16x4_f32(false, a, false, b0,
                                                             (short)0, acc0, false, false);
                acc1 = __builtin_amdgcn_wmma_f32_16x16x4_f32(false, a, false, b1,
                                                             (short)0, acc1, false, false);
                acc2 = __builtin_amdgcn_wmma_f32_16x16x4_f32(false, a, false, b2,
                                                             (short)0, acc2, false, false);
                acc3 = __builtin_amdgcn_wmma_f32_16x16x4_f32(false, a, false, b3,
                                                             (short)0, acc3, false, false);
                continue;
            }
            // Group spans >16 nodes (sparse region): accs are zero here;
            // invalidate window and take the direct-atomic path below.
            win_base = -1;
        }

        // --- Fallback: direct atomic scatter for this (partial/sparse) group ---
#pragma unroll
        for (int i = 0; i < 4; ++i) {
            if (i >= cnt) break;
            const int node = (i == 0) ? c0 : (i == 1) ? c1 : (i == 2) ? c2 : c3;
            const v2f v = *(const v2f*)(edge_feat + (size_t)(base + i) * D_FEAT + lane * 2);
            float* op = out + (size_t)node * D_FEAT + lane * 2;
            atomicAdd(op + 0, v.x);
            atomicAdd(op + 1, v.y);
        }
    }

    if (win_base >= 0) flush_win();
}

extern "C" void kernel_launch(void* const* d_in, const int* in_sizes, int n_in,
                              void* d_out, int out_size, void* d_ws, size_t ws_size,
                              hipStream_t stream) {
    const int*   edge_index = (const int*)d_in[0];   // [2, E]; row 0 = sorted centers
    const float* edge_feat  = (const float*)d_in[1]; // [E, 64]
    const int E = in_sizes[0] / 2;
    const int N = in_sizes[2];                       // node_types is [N, 1]
    float* out = (float*)d_out;                      // [N, 64]

    // Zero the output (harness poisons it; atomics accumulate into zeros).
    const int n4 = out_size / 4;
    er_zero_kernel<<<(n4 + BLOCK - 1) / BLOCK, BLOCK, 0, stream>>>(out, n4);

    const int waves  = (E + EDGES_PER_WAVE - 1) / EDGES_PER_WAVE;
    const int blocks = (waves + WAVES_PER_BLK - 1) / WAVES_PER_BLK;
    er_wmma_scatter_kernel<<<blocks, BLOCK, 0, stream>>>(edge_index, edge_feat, out, E, N);
}